// NeuSField_103079215178
// MI455X (gfx1250) — compile-verified
//
#include <hip/hip_runtime.h>
#include <hip/hip_bf16.h>

typedef __attribute__((ext_vector_type(2))) float v2f;
typedef __attribute__((ext_vector_type(8))) float v8f;
// 4-byte-aligned pair for unaligned-row streaming loads (rows are 35 floats)
typedef float v2fu __attribute__((ext_vector_type(2), aligned(4)));

// D = A(16x4 f32) * B(4x16 f32) + C(16x16 f32), one matrix per wave32.
static __device__ __forceinline__ v8f wmma_f32x4(v2f a, v2f b, v8f c) {
    // 8 args: (neg_a, A, neg_b, B, c_mod, C, reuse_a, reuse_b)
    return __builtin_amdgcn_wmma_f32_16x16x4_f32(
        false, a, false, b, (short)0, c, false, false);
}

// Branchless, stable softplus(beta*h)/beta with hardware transcendentals.
static __device__ __forceinline__ float softplus100(float h) {
    const float z = 100.f * h;
    // softplus(z) = max(z,0) + log(1 + exp(-|z|))
    const float sp = fmaxf(z, 0.f) + __logf(1.f + __expf(-fabsf(z)));
    return sp * 0.01f;
}

// ---------------------------------------------------------------------------
// Prep: fold weight_norm into plain matrices, B-layout [K][N], zero padded.
//   B0[36][64]  : w0^T (k<35 real, k=35 zero)
//   B1[64][16]  : w1^T (n<13 real, n in 13..15 zero)
//   bias1p[16]  : b1 padded with zeros
// ---------------------------------------------------------------------------
__global__ void prep_weights_kernel(const float* __restrict__ W0_v,
                                    const float* __restrict__ W0_g,
                                    const float* __restrict__ W1_v,
                                    const float* __restrict__ W1_g,
                                    const float* __restrict__ b1,
                                    float* __restrict__ B0,
                                    float* __restrict__ B1,
                                    float* __restrict__ bias1p) {
    const int n = threadIdx.x;
    if (n < 64) {
        float s = 0.f;
        for (int k = 0; k < 35; ++k) { float v = W0_v[n * 35 + k]; s += v * v; }
        const float inv = W0_g[n] * rsqrtf(s);
        for (int k = 0; k < 35; ++k) B0[k * 64 + n] = W0_v[n * 35 + k] * inv;
        B0[35 * 64 + n] = 0.f;
    }
    if (n < 16) {
        if (n < 13) {
            float s = 0.f;
            for (int k = 0; k < 64; ++k) { float v = W1_v[n * 64 + k]; s += v * v; }
            const float inv = W1_g[n] * rsqrtf(s);
            for (int k = 0; k < 64; ++k) B1[k * 16 + n] = W1_v[n * 64 + k] * inv;
            bias1p[n] = b1[n];
        } else {
            for (int k = 0; k < 64; ++k) B1[k * 16 + n] = 0.f;
            bias1p[n] = 0.f;
        }
    }
}

// ---------------------------------------------------------------------------
// Main: each wave processes one 16-row tile per grid-stride step.
//   layer0: 9 K-chunks x 4 N-chunks of f32 WMMA (K padded 35->36)
//   branchless softplus, re-swizzle via per-wave LDS slab (stride 68)
//   layer1: 16 K-chunks x 1 N-chunk, predicated NT store of 13 columns
// ---------------------------------------------------------------------------
__global__ __launch_bounds__(256) void mlp_wmma_kernel(
    const float* __restrict__ x,
    const float* __restrict__ b0,
    const float* __restrict__ B0,
    const float* __restrict__ B1,
    const float* __restrict__ bias1,
    float* __restrict__ out,
    int N) {
    __shared__ float hbuf[8][16][68];   // per-wave slab, stride 68 -> no bank conflicts

    const int lane = threadIdx.x & 31;
    const int wave = threadIdx.x >> 5;
    const int lrow = lane & 15;         // N (or M) index within half-wave
    const int lhi  = lane >> 4;         // upper half-wave: +2 K (A/B), +8 M (C/D)

    // ---- preload all B fragments into registers (reused every tile) ----
    v2f B0f[9][4];
#pragma unroll
    for (int q = 0; q < 9; ++q) {
#pragma unroll
        for (int c = 0; c < 4; ++c) {
            const int k0  = 4 * q + 2 * lhi;
            const int col = 16 * c + lrow;
            v2f t;
            t.x = B0[k0 * 64 + col];
            t.y = B0[(k0 + 1) * 64 + col];
            B0f[q][c] = t;
        }
    }
    v2f B1f[16];
#pragma unroll
    for (int q = 0; q < 16; ++q) {
        const int k0 = 4 * q + 2 * lhi;
        v2f t;
        t.x = B1[k0 * 16 + lrow];
        t.y = B1[(k0 + 1) * 16 + lrow];
        B1f[q] = t;
    }
    float bias0c[4];
#pragma unroll
    for (int c = 0; c < 4; ++c) bias0c[c] = b0[16 * c + lrow];
    const float bias1v = bias1[lrow];

    const long ntiles  = ((long)N + 15) >> 4;
    const long wstride = (long)gridDim.x * 8;

    for (long t = (long)blockIdx.x * 8 + wave; t < ntiles; t += wstride) {
        long row = t * 16 + lrow;
        if (row >= N) row = N - 1;                    // clamp loads, stores predicated
        const float* xr = x + row * 35;

        // prefetch next tile's rows for this lane
        const long tn = t + wstride;
        if (tn < ntiles) {
            long prow = tn * 16 + lrow;
            if (prow >= N) prow = N - 1;
            __builtin_prefetch(x + prow * 35, 0, 1);
        }

        // ---- A fragments for layer 0 (nontemporal stream, branchless pad) ----
        // lane covers K = {k0, k0+1}, k0 = 4q + 2*lhi
        v2f A0[9];
#pragma unroll
        for (int q = 0; q < 8; ++q) {
            const int k0 = 4 * q + 2 * lhi;
            v2fu av = __builtin_nontemporal_load((const v2fu*)(xr + k0));
            v2f a; a.x = av.x; a.y = av.y;
            A0[q] = a;
        }
        {   // K chunk 8 covers k = 32..35; element 35 is the zero pad.
            // All lanes load valid addresses; pad selected via cndmask (no branch).
            const float e_lo = __builtin_nontemporal_load(xr + 32 + 2 * lhi); // 32 or 34
            const float e33  = __builtin_nontemporal_load(xr + 33);
            v2f a;
            a.x = e_lo;
            a.y = lhi ? 0.f : e33;
            A0[8] = a;
        }

        // ---- layer 0: C init = b0 broadcast, then 36 WMMAs ----
        v8f C0[4];
#pragma unroll
        for (int c = 0; c < 4; ++c) {
#pragma unroll
            for (int r = 0; r < 8; ++r) C0[c][r] = bias0c[c];
        }
#pragma unroll
        for (int c = 0; c < 4; ++c) {
#pragma unroll
            for (int q = 0; q < 9; ++q) {
                C0[c] = wmma_f32x4(A0[q], B0f[q][c], C0[c]);
            }
        }

        // ---- branchless softplus, write h tile to LDS (C-layout -> row major) ----
#pragma unroll
        for (int c = 0; c < 4; ++c) {
#pragma unroll
            for (int r = 0; r < 8; ++r) {
                hbuf[wave][r + 8 * lhi][16 * c + lrow] = softplus100(C0[c][r]);
            }
        }

        // ---- layer 1: read A fragments back (A-layout), 16 WMMAs ----
        v8f C1;
#pragma unroll
        for (int r = 0; r < 8; ++r) C1[r] = bias1v;
#pragma unroll
        for (int q = 0; q < 16; ++q) {
            const float* hp = &hbuf[wave][lrow][4 * q + 2 * lhi];
            v2f a; a.x = hp[0]; a.y = hp[1];
            C1 = wmma_f32x4(a, B1f[q], C1);
        }

        // ---- store: N=lrow (<13), M = r + 8*lhi, nontemporal stream ----
        if (lrow < 13) {
#pragma unroll
            for (int r = 0; r < 8; ++r) {
                const long orow = t * 16 + r + 8 * lhi;
                if (orow < N) __builtin_nontemporal_store(C1[r], &out[orow * 13 + lrow]);
            }
        }
    }
}

extern "C" void kernel_launch(void* const* d_in, const int* in_sizes, int n_in,
                              void* d_out, int out_size, void* d_ws, size_t ws_size,
                              hipStream_t stream) {
    const float* x    = (const float*)d_in[0];
    const float* W0_v = (const float*)d_in[1];
    const float* W0_g = (const float*)d_in[2];
    const float* b0   = (const float*)d_in[3];
    const float* W1_v = (const float*)d_in[4];
    const float* W1_g = (const float*)d_in[5];
    const float* b1   = (const float*)d_in[6];
    float* out = (float*)d_out;

    const int N = in_sizes[0] / 35;

    float* wsf    = (float*)d_ws;
    float* B0     = wsf;                 // 36*64 floats
    float* B1     = B0 + 36 * 64;        // 64*16 floats
    float* bias1p = B1 + 64 * 16;        // 16 floats

    prep_weights_kernel<<<1, 64, 0, stream>>>(W0_v, W0_g, W1_v, W1_g, b1, B0, B1, bias1p);

    const long tiles = ((long)N + 15) / 16;
    int blocks = (int)((tiles + 7) / 8);
    if (blocks > 2048) blocks = 2048;
    if (blocks < 1) blocks = 1;
    mlp_wmma_kernel<<<blocks, 256, 0, stream>>>(x, b0, B0, B1, bias1p, out, N);
}